// SemanticAwarePromptLearner_86560770884266
// MI455X (gfx1250) — compile-verified
//
#include <hip/hip_runtime.h>
#include <hip/hip_bf16.h>

typedef __attribute__((ext_vector_type(2))) float v2f;
typedef __attribute__((ext_vector_type(4))) float v4f;
typedef __attribute__((ext_vector_type(8))) float v8f;

#define Bc 32
#define Pc 196
#define Dc 512
#define Hc 8
#define Uc 4
#define Cc 500

// ---------------------------------------------------------------------------
// Kernel A: scores -> softmax(P) -> semantic_tokens via V_WMMA_F32_16X16X4_F32
// One block (256 threads = 8 waves) per batch element.
// Weights tile in LDS is padded to M=16 (rows 8..15 zero) so the A-fragment
// loads in the WMMA loop are unconditional (no exec-mask juggling).
// ---------------------------------------------------------------------------
__global__ __launch_bounds__(256) void attn_kernel(const float* __restrict__ patch,
                                                   const float* __restrict__ query,
                                                   float* __restrict__ sem_out) {
    __shared__ float qs[Hc * Dc];        // 16 KB: all 8 query rows
    __shared__ float w16[Pc * 16];       // 12.25 KB: softmax weights, M padded to 16
    const int b   = blockIdx.x;
    const int tid = threadIdx.x;
    const float* pb = patch + (size_t)b * Pc * Dc;

    // Stage 0: queries into LDS (coalesced 16B) + zero-fill padded weight tile
    for (int i = tid; i < (Hc * Dc) / 4; i += 256)
        ((v4f*)qs)[i] = ((const v4f*)query)[i];
    for (int i = tid; i < (Pc * 16) / 4; i += 256)
        ((v4f*)w16)[i] = (v4f){0.f, 0.f, 0.f, 0.f};
    __syncthreads();

    // Stage 1: scores[p][h] = dot(patch[b,p,:], Q[h,:])  (one p per thread)
    if (tid < Pc) {
        float acc[Hc] = {};
        for (int d = 0; d < Dc; d += 4) {
            const v4f pv = *(const v4f*)(pb + (size_t)tid * Dc + d);
#pragma unroll
            for (int h = 0; h < Hc; ++h) {
                const v4f qv = *(const v4f*)(qs + h * Dc + d);
                acc[h] += pv.x * qv.x + pv.y * qv.y + pv.z * qv.z + pv.w * qv.w;
            }
        }
#pragma unroll
        for (int h = 0; h < Hc; ++h) w16[tid * 16 + h] = acc[h];
    }
    __syncthreads();

    // Stage 2: softmax over P for each head (8 heads; rows 8..15 stay zero)
    if (tid < Hc) {
        float mx = -3.402823e38f;
        for (int p = 0; p < Pc; ++p) mx = fmaxf(mx, w16[p * 16 + tid]);
        float s = 0.f;
        for (int p = 0; p < Pc; ++p) {
            const float e = __expf(w16[p * 16 + tid] - mx);
            w16[p * 16 + tid] = e;
            s += e;
        }
        const float inv = 1.f / s;
        for (int p = 0; p < Pc; ++p) w16[p * 16 + tid] *= inv;
    }
    __syncthreads();

    // Stage 3: semantic[h][d] = sum_p w[p][h] * patch[p][d]
    // WMMA: A = Wt (M=16 padded, K=196), B = patch (K=196, N=16 tile), 32 N-tiles.
    const int wave = tid >> 5;
    const int lane = tid & 31;
    const int ml   = lane & 15;           // M for A / N for B / N for C
    const int kb   = (lane >> 4) * 2;     // K sub-offset per half-wave
    for (int nt = wave; nt < Dc / 16; nt += 8) {
        const int nbase = nt * 16;
        v8f c = {};
        for (int k0 = 0; k0 < Pc; k0 += 4) {
            v2f a, bf;
            a.x  = w16[(k0 + kb) * 16 + ml];
            a.y  = w16[(k0 + kb + 1) * 16 + ml];
            bf.x = pb[(size_t)(k0 + kb) * Dc + nbase + ml];
            bf.y = pb[(size_t)(k0 + kb + 1) * Dc + nbase + ml];
            c = __builtin_amdgcn_wmma_f32_16x16x4_f32(false, a, false, bf,
                                                      (short)0, c, false, false);
        }
        if (lane < 16) {                  // lanes 0-15 hold M=0..7 -> h rows
#pragma unroll
            for (int r = 0; r < Hc; ++r)
                sem_out[(size_t)b * Hc * Dc + r * Dc + nbase + ml] = c[r];
        }
    }
}

// ---------------------------------------------------------------------------
// Kernel B: projected_domain = gf @ dom_W^T + dom_b   (M=32, N=512, K=512)
// One wave per 16x16 tile; grid = (Ntiles=32, Mtiles=2). A/B fragment fetches
// are contiguous v2f -> single global_load_b64 per fragment half.
// ---------------------------------------------------------------------------
__global__ __launch_bounds__(32) void dom_kernel(const float* __restrict__ gf,
                                                 const float* __restrict__ W,
                                                 const float* __restrict__ bias,
                                                 float* __restrict__ out) {
    const int nbase = blockIdx.x * 16;
    const int mbase = blockIdx.y * 16;
    const int lane  = threadIdx.x;
    const int ml    = lane & 15;
    const int kb    = (lane >> 4) * 2;
    const float* arow = gf + (size_t)(mbase + ml) * Dc + kb;
    const float* brow = W  + (size_t)(nbase + ml) * Dc + kb;
    v8f c = {};
#pragma unroll 4
    for (int k0 = 0; k0 < Dc; k0 += 4) {
        const v2f a  = *(const v2f*)(arow + k0);
        const v2f bf = *(const v2f*)(brow + k0);
        c = __builtin_amdgcn_wmma_f32_16x16x4_f32(false, a, false, bf,
                                                  (short)0, c, false, false);
    }
    const int row0 = mbase + (lane >> 4) * 8;
#pragma unroll
    for (int r = 0; r < 8; ++r)
        out[(size_t)(row0 + r) * Dc + nbase + ml] = c[r] + bias[nbase + ml];
}

// ---------------------------------------------------------------------------
// Kernel C: projected_sem[b,h,e] = sum_d sem[b,h,d]*sem_W[h,e,d] + sem_b[h,e]
// Per h: M=32, N=512, K=512. grid = (Ntiles=32, Mtiles=2, H=8)
// ---------------------------------------------------------------------------
__global__ __launch_bounds__(32) void semproj_kernel(const float* __restrict__ sem,
                                                     const float* __restrict__ W,
                                                     const float* __restrict__ bias,
                                                     float* __restrict__ out) {
    const int nbase = blockIdx.x * 16;
    const int mbase = blockIdx.y * 16;
    const int h     = blockIdx.z;
    const int lane  = threadIdx.x;
    const int ml    = lane & 15;
    const int kb    = (lane >> 4) * 2;
    const float* arow = sem + ((size_t)(mbase + ml) * Hc + h) * Dc + kb;
    const float* brow = W   + ((size_t)h * Dc + nbase + ml) * Dc + kb;
    v8f c = {};
#pragma unroll 4
    for (int k0 = 0; k0 < Dc; k0 += 4) {
        const v2f a  = *(const v2f*)(arow + k0);
        const v2f bf = *(const v2f*)(brow + k0);
        c = __builtin_amdgcn_wmma_f32_16x16x4_f32(false, a, false, bf,
                                                  (short)0, c, false, false);
    }
    const int row0 = mbase + (lane >> 4) * 8;
#pragma unroll
    for (int r = 0; r < 8; ++r)
        out[((size_t)(row0 + r) * Hc + h) * Dc + nbase + ml] =
            c[r] + bias[(size_t)h * Dc + nbase + ml];
}

// ---------------------------------------------------------------------------
// Kernel D: assemble known_prompts (b*c, 11, 512) -- 360 MB of NT b128 stores.
// grid = B*C, 128 threads; each thread stores one 16B vector per row.
// ---------------------------------------------------------------------------
__global__ __launch_bounds__(128) void known_kernel(const float* __restrict__ prefix,
                                                    const float* __restrict__ suffix,
                                                    const float* __restrict__ dom,
                                                    const float* __restrict__ psem,
                                                    float* __restrict__ out) {
    const int bc = blockIdx.x;
    const int b  = bc / Cc;
    const int c  = bc % Cc;
    const int t  = threadIdx.x;
    float* o = out + (size_t)bc * 11 * Dc;

    const float* srcs[11];
    srcs[0] = prefix + (size_t)c * Dc;
    srcs[1] = dom + (size_t)b * Dc;
#pragma unroll
    for (int h = 0; h < Hc; ++h) srcs[2 + h] = psem + ((size_t)b * Hc + h) * Dc;
    srcs[10] = suffix + (size_t)c * Dc;

    // warm the gathered (c-indexed) rows; b-indexed rows are L2-resident anyway
    __builtin_prefetch(srcs[0] + t * 4, 0, 3);
    __builtin_prefetch(srcs[10] + t * 4, 0, 3);

#pragma unroll
    for (int row = 0; row < 11; ++row) {
        const v4f v = *(const v4f*)(srcs[row] + t * 4);
        __builtin_nontemporal_store(v, (v4f*)(o + row * Dc + t * 4));
    }
}

// ---------------------------------------------------------------------------
// Kernel E: unknown_prompts (32, 7, 512)
// ---------------------------------------------------------------------------
__global__ __launch_bounds__(128) void unk_kernel(const float* __restrict__ upre,
                                                  const float* __restrict__ usuf,
                                                  const float* __restrict__ dom,
                                                  const float* __restrict__ utok,
                                                  float* __restrict__ out) {
    const int b = blockIdx.x;
    const int t = threadIdx.x;
    float* o = out + (size_t)b * 7 * Dc;
    const float* srcs[7] = { upre, dom + (size_t)b * Dc,
                             utok, utok + Dc, utok + 2 * Dc, utok + 3 * Dc,
                             usuf };
#pragma unroll
    for (int row = 0; row < 7; ++row) {
        const v4f v = *(const v4f*)(srcs[row] + t * 4);
        __builtin_nontemporal_store(v, (v4f*)(o + row * Dc + t * 4));
    }
}

extern "C" void kernel_launch(void* const* d_in, const int* in_sizes, int n_in,
                              void* d_out, int out_size, void* d_ws, size_t ws_size,
                              hipStream_t stream) {
    const float* patch  = (const float*)d_in[0];
    const float* gf     = (const float*)d_in[1];
    const float* query  = (const float*)d_in[2];
    const float* dom_W  = (const float*)d_in[3];
    const float* dom_b  = (const float*)d_in[4];
    const float* sem_W  = (const float*)d_in[5];
    const float* sem_b  = (const float*)d_in[6];
    const float* utok   = (const float*)d_in[7];
    const float* kpre   = (const float*)d_in[8];
    const float* ksuf   = (const float*)d_in[9];
    const float* upre   = (const float*)d_in[10];
    const float* usuf   = (const float*)d_in[11];

    float* out = (float*)d_out;
    const size_t known_elems = (size_t)Bc * Cc * (Hc + 3) * Dc;  // 90,112,000
    const size_t unk_elems   = (size_t)Bc * (Uc + 3) * Dc;       //    114,688
    float* known   = out;
    float* unk     = out + known_elems;
    float* sem_tok = out + known_elems + unk_elems;              // 3rd output

    float* dom  = (float*)d_ws;                                  // 32*512
    float* psem = dom + (size_t)Bc * Dc;                         // 32*8*512

    attn_kernel<<<Bc, 256, 0, stream>>>(patch, query, sem_tok);
    dom_kernel<<<dim3(Dc / 16, Bc / 16), 32, 0, stream>>>(gf, dom_W, dom_b, dom);
    semproj_kernel<<<dim3(Dc / 16, Bc / 16, Hc), 32, 0, stream>>>(sem_tok, sem_W, sem_b, psem);
    known_kernel<<<Bc * Cc, 128, 0, stream>>>(kpre, ksuf, dom, psem, known);
    unk_kernel<<<Bc, 128, 0, stream>>>(upre, usuf, dom, utok, unk);
}